// CustomModel_71227737637113
// MI455X (gfx1250) — compile-verified
//
#include <hip/hip_runtime.h>
#include <math.h>

typedef float v2f __attribute__((ext_vector_type(2)));
typedef float v8f __attribute__((ext_vector_type(8)));

#define BATCH 64
#define NPTS  128
#define DIM   64

// ---------------------------------------------------------------------------
// Kernel 1: dist[b,i,j] = sqrt(||t_i||^2 + ||p_j||^2 - 2 t_i.p_j)
// One wave32 per 16x16 output tile, FP32 WMMA 16x16x4, K=64 -> 16 WMMA ops.
// ---------------------------------------------------------------------------
__global__ __launch_bounds__(128) void dist_wmma_kernel(
    const float* __restrict__ T, const float* __restrict__ P,
    float* __restrict__ dist, float* __restrict__ out)
{
  if (blockIdx.x == 0 && threadIdx.x == 0) out[0] = 0.0f; // harness poisons d_out

  const int wave = threadIdx.x >> 5;
  const int lane = threadIdx.x & 31;
  const int tile = blockIdx.x * 4 + wave;   // 0 .. BATCH*64-1
  const int b  = tile >> 6;
  const int t  = tile & 63;
  const int ti = t >> 3;                    // row tile 0..7
  const int tj = t & 7;                     // col tile 0..7

  const float* Tb = T + (size_t)b * NPTS * DIM;
  const float* Pb = P + (size_t)b * NPTS * DIM;

  const int m  = lane & 15;                 // matrix row (A) / col (B) index
  const int kh = lane >> 4;                 // K-half select (0 -> K0/K1, 1 -> K2/K3)

  // A frag: 16x4 fp32, lane m holds row ti*16+m, VGPR v = K (kh*2+v)
  // B frag: 4x16 fp32 = P^T tile, lane n holds col tj*16+n, VGPR v = K (kh*2+v)
  v8f acc = {0.f,0.f,0.f,0.f,0.f,0.f,0.f,0.f};
  const float* arow = Tb + (size_t)(ti * 16 + m) * DIM + kh * 2;
  const float* brow = Pb + (size_t)(tj * 16 + m) * DIM + kh * 2;
#pragma unroll
  for (int k0 = 0; k0 < DIM; k0 += 4) {
    v2f a  = *(const v2f*)(arow + k0);
    v2f bb = *(const v2f*)(brow + k0);
    acc = __builtin_amdgcn_wmma_f32_16x16x4_f32(false, a, false, bb,
                                                (short)0, acc, false, false);
  }

  // Squared norms: lanes 0-15 -> rows of T tile, lanes 16-31 -> rows of P tile
  float nrm = 0.f;
  const float* nr = (lane < 16) ? (Tb + (size_t)(ti * 16 + lane) * DIM)
                                : (Pb + (size_t)(tj * 16 + (lane - 16)) * DIM);
#pragma unroll
  for (int k = 0; k < DIM; k += 4) {
    float4 q = *(const float4*)(nr + k);
    nrm += q.x * q.x + q.y * q.y + q.z * q.z + q.w * q.w;
  }

  // C/D layout: VGPR v, lanes 0-15 -> M=v, lanes 16-31 -> M=v+8, N=lane&15
  float* Db = dist + (size_t)b * NPTS * NPTS;
  const int n = lane & 15;
  const float npn = __shfl(nrm, 16 + n);    // ||p_{tj*16+n}||^2
#pragma unroll
  for (int v = 0; v < 8; ++v) {
    const int mrow = v + 8 * kh;
    const float ntm = __shfl(nrm, mrow);    // ||t_{ti*16+mrow}||^2
    const float d2 = ntm + npn - 2.0f * acc[v];
    Db[(size_t)(ti * 16 + mrow) * NPTS + tj * 16 + n] = sqrtf(fmaxf(d2, 0.0f));
  }
}

// ---------------------------------------------------------------------------
// Kernel 2: Jonker-Volgenant Hungarian, one workgroup (128 threads) per batch.
// Thread tid owns column (tid+1); column 0 is the virtual start column.
// ---------------------------------------------------------------------------
__global__ __launch_bounds__(NPTS) void hungarian_kernel(
    const float* __restrict__ dist, float* __restrict__ out)
{
  __shared__ float u[NPTS + 1], vv[NPTS + 1], minv[NPTS + 1];
  __shared__ int   p[NPTS + 1], way[NPTS + 1];
  __shared__ unsigned char used[NPTS + 1];
  __shared__ float redv[NPTS];
  __shared__ int   redi[NPTS];
  __shared__ int   j0s;

  const int tid = threadIdx.x;              // 0..127
  const int j   = tid + 1;                  // owned column (1-indexed)
  const float* C = dist + (size_t)blockIdx.x * NPTS * NPTS;

  u[j] = 0.f; vv[j] = 0.f; p[j] = 0; way[j] = 0;
  if (tid == 0) { u[0] = 0.f; vv[0] = 0.f; p[0] = 0; way[0] = 0; }
  __syncthreads();

  for (int i = 1; i <= NPTS; ++i) {
    minv[j] = 1e30f; used[j] = 0;
    if (tid == 0) { used[0] = 0; p[0] = i; j0s = 0; }
    __syncthreads();

    while (true) {
      const int j0 = j0s;
      if (tid == 0) used[j0] = 1;
      __syncthreads();

      const int   i0  = p[j0];
      const float ui0 = u[i0];
      float mv = minv[j];
      if (!used[j]) {
        const float cur = C[(size_t)(i0 - 1) * NPTS + tid] - ui0 - vv[j];
        if (cur < mv) { mv = cur; minv[j] = cur; way[j] = j0; }
      }
      redv[tid] = used[j] ? 1e30f : mv;
      redi[tid] = j;
      __syncthreads();
      for (int s = NPTS / 2; s > 0; s >>= 1) {
        if (tid < s && redv[tid + s] < redv[tid]) {
          redv[tid] = redv[tid + s];
          redi[tid] = redi[tid + s];
        }
        __syncthreads();
      }
      const float delta = redv[0];
      const int   j1    = redi[0];
      __syncthreads();

      // Dual update: used cols shift duals, unused cols shrink minv.
      if (used[j]) { vv[j] -= delta; u[p[j]] += delta; }
      else         { minv[j] -= delta; }
      if (tid == 0) { vv[0] -= delta; u[p[0]] += delta; j0s = j1; }
      __syncthreads();

      if (p[j1] == 0) break;                // found an unmatched column
    }

    // Augment along the alternating path (sequential, thread 0)
    if (tid == 0) {
      int jj = j0s;
      while (jj) { const int jp = way[jj]; p[jj] = p[jp]; jj = jp; }
    }
    __syncthreads();
  }

  // Matched sum: column j is matched to row p[j]-1
  redv[tid] = C[(size_t)(p[j] - 1) * NPTS + tid];
  __syncthreads();
  for (int s = NPTS / 2; s > 0; s >>= 1) {
    if (tid < s) redv[tid] += redv[tid + s];
    __syncthreads();
  }
  if (tid == 0) atomicAdd(out, redv[0] * (1.0f / BATCH));
}

// ---------------------------------------------------------------------------
extern "C" void kernel_launch(void* const* d_in, const int* in_sizes, int n_in,
                              void* d_out, int out_size, void* d_ws, size_t ws_size,
                              hipStream_t stream) {
  const float* y_true = (const float*)d_in[0];  // [B, N, D] fp32
  const float* y_pred = (const float*)d_in[1];  // [B, N, D] fp32
  float* out  = (float*)d_out;                  // scalar fp32
  float* dist = (float*)d_ws;                   // B*N*N fp32 = 4 MB scratch

  // 4096 tiles, 4 waves (tiles) per 128-thread block -> 1024 blocks
  dist_wmma_kernel<<<dim3(BATCH * 64 / 4), 128, 0, stream>>>(y_true, y_pred, dist, out);
  hungarian_kernel<<<dim3(BATCH), NPTS, 0, stream>>>(dist, out);
}